// MSREnhancement_72000831750411
// MI455X (gfx1250) — compile-verified
//
#include <hip/hip_runtime.h>
#include <hip/hip_bf16.h>

typedef __attribute__((ext_vector_type(2))) float v2f;
typedef __attribute__((ext_vector_type(8))) float v8f;

#define NPLANE 24      // 8 batch * 3 channels
#define HFULL 1280
#define EPSV 1e-6f

__device__ __forceinline__ int refl101(int i, int n) {
  if (i < 0) i = -i;
  if (i >= n) i = 2 * n - 2 - i;
  if (i < 0) i = 0;            // safety clamp for deep OOB on tiny levels
  if (i >= n) i = n - 1;
  return i;
}

// 5-tap binomial weight [1,4,6,4,1]/16, zero outside 0..4
__device__ __forceinline__ float binw(int d) {
  if (d == 0 || d == 4) return 0.0625f;
  if (d == 1 || d == 3) return 0.25f;
  if (d == 2) return 0.375f;
  return 0.0f;
}

// ---------------------------------------------------------------------------
// pyrDown (cv2 semantics: 5x5 binomial, reflect-101, stride-2).
// One wave32 per 16x16 output tile.
// Vertical 5-tap+decimate -> V(16x36) in LDS (col 35 zero pad).
// Horizontal+decimate as banded matmul on the matrix core:
//   Out(16x16) = V(16x36) x Hmat(36x16),  Hmat[c][j] = binw(c - 2j)
// accumulated with 9 chained V_WMMA_F32_16X16X4_F32 (full f32 precision).
// ---------------------------------------------------------------------------
__global__ __launch_bounds__(32) void pyrdown_wmma(
    const float* __restrict__ src, float* __restrict__ dst,
    int W, int outW, float scale, float bias) {
  __shared__ float sIn[35 * 36];
  __shared__ float sV[16 * 36];
  const int tid = threadIdx.x;
  const int ox0 = blockIdx.x * 16;
  const int oy0 = blockIdx.y * 16;
  const int plane = blockIdx.z;
  const float* sp = src + (size_t)plane * W * W;

  // Stage 35x35 input region (reflect-101) into LDS, fusing img = scale*x+bias.
  for (int idx = tid; idx < 35 * 35; idx += 32) {
    int r = idx / 35, c = idx % 35;
    int gr = refl101(2 * oy0 - 2 + r, W);
    int gc = refl101(2 * ox0 - 2 + c, W);
    sIn[r * 36 + c] = scale * sp[gr * W + gc] + bias;
  }
  __syncthreads();

  // Vertical 5-tap + row decimation: V[i][c], i=0..15 out rows, c=0..34 cols.
  for (int idx = tid; idx < 16 * 36; idx += 32) {
    int i = idx / 36, c = idx % 36;
    float v = 0.0f;
    if (c < 35) {
      v = 0.0625f * sIn[(2 * i + 0) * 36 + c] + 0.25f * sIn[(2 * i + 1) * 36 + c]
        + 0.375f  * sIn[(2 * i + 2) * 36 + c] + 0.25f * sIn[(2 * i + 3) * 36 + c]
        + 0.0625f * sIn[(2 * i + 4) * 36 + c];
    }
    sV[idx] = v;
  }
  __syncthreads();

  // A fragment (16x4 f32): lanes 0-15 hold K0/K1, lanes 16-31 hold K2/K3.
  // B fragment (4x16 f32): lanes 0-15 rows K0/K1 (N=lane), lanes 16-31 rows K2/K3.
  const int half = tid >> 4;
  const int lm = tid & 15;
  v8f acc = {};
#pragma unroll
  for (int q = 0; q < 9; ++q) {
    int k0 = 4 * q + 2 * half;
    v2f a, b;
    a.x = sV[lm * 36 + k0];
    a.y = sV[lm * 36 + k0 + 1];
    int d0 = k0 - 2 * lm;             // band offset into binomial kernel
    b.x = binw(d0);
    b.y = binw(d0 + 1);
    acc = __builtin_amdgcn_wmma_f32_16x16x4_f32(false, a, false, b,
                                                (short)0, acc, false, false);
  }

  // D layout: VGPR v -> row v (lanes 0-15) / row v+8 (lanes 16-31), col = lane%16.
  float* dp = dst + (size_t)plane * outW * outW;
#pragma unroll
  for (int v = 0; v < 8; ++v) {
    int row = oy0 + v + 8 * half;
    int col = ox0 + lm;
    if (row < outW && col < outW) dp[row * outW + col] = acc[v];
  }
}

// ---------------------------------------------------------------------------
// Small-level Gaussian blur (cv2.GaussianBlur, sigma from ksize), reflect-101.
// Only runs on 40x40 / 10x10 / 5x5 levels -> negligible cost, keep it simple.
// ---------------------------------------------------------------------------
__global__ void gauss_small(const float* __restrict__ src, float* __restrict__ dst,
                            int w, int ksize, float sig) {
  int tid = blockIdx.x * blockDim.x + threadIdx.x;
  int tot = NPLANE * w * w;
  if (tid >= tot) return;
  int plane = tid / (w * w);
  int rem = tid - plane * w * w;
  int y = rem / w, x = rem - y * w;

  float g[9];
  float norm = 0.0f;
  float ctr = (ksize - 1) * 0.5f;
  for (int i = 0; i < ksize; ++i) {
    float d = (float)i - ctr;
    g[i] = expf(-(d * d) / (2.0f * sig * sig));
    norm += g[i];
  }
  const float* sp = src + (size_t)plane * w * w;
  int p = ksize / 2;
  float acc = 0.0f;
  for (int ky = 0; ky < ksize; ++ky) {
    int yy = refl101(y + ky - p, w);
    float rowacc = 0.0f;
    for (int kx = 0; kx < ksize; ++kx) {
      int xx = refl101(x + kx - p, w);
      rowacc += g[kx] * sp[yy * w + xx];
    }
    acc += g[ky] * rowacc;
  }
  dst[tid] = acc / (norm * norm);
}

// ---------------------------------------------------------------------------
// Exact-2x half-pixel bilinear upsample (jax.image.resize 'bilinear', upscale).
// even out y=2j: 0.25*S[j-1] + 0.75*S[j] ; odd y=2j+1: 0.75*S[j] + 0.25*S[j+1]
// ---------------------------------------------------------------------------
__global__ void upsample2x(const float* __restrict__ src, float* __restrict__ dst, int w) {
  int W2 = 2 * w;
  int tid = blockIdx.x * blockDim.x + threadIdx.x;
  int tot = NPLANE * W2 * W2;
  if (tid >= tot) return;
  int plane = tid / (W2 * W2);
  int rem = tid - plane * W2 * W2;
  int y = rem / W2, x = rem - y * W2;

  int jy = y >> 1, jx = x >> 1;
  int y0; float fy;
  if (y & 1) { y0 = jy; fy = 0.25f; } else { y0 = jy - 1; fy = 0.75f; }
  int x0; float fx;
  if (x & 1) { x0 = jx; fx = 0.25f; } else { x0 = jx - 1; fx = 0.75f; }
  int y1 = y0 + 1, x1 = x0 + 1;
  if (y0 < 0) y0 = 0;  if (y1 > w - 1) y1 = w - 1;
  if (x0 < 0) x0 = 0;  if (x1 > w - 1) x1 = w - 1;

  const float* sp = src + (size_t)plane * w * w;
  float v00 = sp[y0 * w + x0], v01 = sp[y0 * w + x1];
  float v10 = sp[y1 * w + x0], v11 = sp[y1 * w + x1];
  float a = v00 + fx * (v01 - v00);
  float c = v10 + fx * (v11 - v10);
  dst[tid] = a + fy * (c - a);
}

// ---------------------------------------------------------------------------
// Final fused pass: reads x + three 640^2 blur maps, does the last 2x bilerp
// inline (saves materializing 3 full-res maps), retinex + color restoration.
// One thread per (n,y,x), handles all 3 channels (shares img_sum + weights).
// ---------------------------------------------------------------------------
__device__ __forceinline__ float bilerp640(const float* __restrict__ b,
                                           int y0, int y1, int x0, int x1,
                                           float fy, float fx) {
  float v00 = b[y0 * 640 + x0], v01 = b[y0 * 640 + x1];
  float v10 = b[y1 * 640 + x0], v11 = b[y1 * 640 + x1];
  float a = v00 + fx * (v01 - v00);
  float c = v10 + fx * (v11 - v10);
  return a + fy * (c - a);
}

__global__ void final_msr(const float* __restrict__ xin,
                          const float* __restrict__ b30,
                          const float* __restrict__ b150,
                          const float* __restrict__ b300,
                          float* __restrict__ out) {
  const int HW = HFULL * HFULL;
  int tid = blockIdx.x * blockDim.x + threadIdx.x;
  if (tid >= 8 * HW) return;
  int n = tid / HW;
  int rem = tid - n * HW;
  int y = rem / HFULL, x = rem - y * HFULL;

  // half-pixel 2x bilerp coords into the 640 grid
  int jy = y >> 1, jx = x >> 1;
  int y0; float fy;
  if (y & 1) { y0 = jy; fy = 0.25f; } else { y0 = jy - 1; fy = 0.75f; }
  int x0; float fx;
  if (x & 1) { x0 = jx; fx = 0.25f; } else { x0 = jx - 1; fx = 0.75f; }
  int y1 = y0 + 1, x1 = x0 + 1;
  if (y0 < 0) y0 = 0;  if (y1 > 639) y1 = 639;
  if (x0 < 0) x0 = 0;  if (x1 > 639) x1 = 639;

  float img[3];
#pragma unroll
  for (int c = 0; c < 3; ++c)
    img[c] = xin[((size_t)(n * 3 + c)) * HW + rem] * 255.0f + 1.0f;
  float s = img[0] + img[1] + img[2];

#pragma unroll
  for (int c = 0; c < 3; ++c) {
    size_t base = (size_t)(n * 3 + c) * 640 * 640;
    float g30  = bilerp640(b30  + base, y0, y1, x0, x1, fy, fx);
    float g150 = bilerp640(b150 + base, y0, y1, x0, x1, fy, fx);
    float g300 = bilerp640(b300 + base, y0, y1, x0, x1, fy, fx);
    float li = log10f(img[c] + EPSV);
    float r = (li - log10f(g30 + EPSV)) + (li - log10f(g150 + EPSV))
            + (li - log10f(g300 + EPSV));
    r *= (1.0f / 3.0f);
    float cr = log10f(img[c] * 2.0f / (s + EPSV) + 1.0f);
    float e = r * cr * 2700.0f + 128.0f;
    e = fminf(fmaxf(e, 0.0f), 255.0f) * (1.0f / 255.0f);
    out[((size_t)(n * 3 + c)) * HW + rem] = e;
  }
}

// ---------------------------------------------------------------------------
extern "C" void kernel_launch(void* const* d_in, const int* in_sizes, int n_in,
                              void* d_out, int out_size, void* d_ws, size_t ws_size,
                              hipStream_t stream) {
  const float* x = (const float*)d_in[0];
  float* out = (float*)d_out;
  float* ws = (float*)d_ws;

  size_t off = 0;
  auto alloc = [&](size_t nfloats) { float* p = ws + off; off += nfloats; return p; };
  float* p1 = alloc((size_t)NPLANE * 640 * 640);  // pyramid L1..L8
  float* p2 = alloc((size_t)NPLANE * 320 * 320);
  float* p3 = alloc((size_t)NPLANE * 160 * 160);
  float* p4 = alloc((size_t)NPLANE * 80 * 80);
  float* p5 = alloc((size_t)NPLANE * 40 * 40);
  float* p6 = alloc((size_t)NPLANE * 20 * 20);
  float* p7 = alloc((size_t)NPLANE * 10 * 10);
  float* p8 = alloc((size_t)NPLANE * 5 * 5);
  float* b30  = alloc((size_t)NPLANE * 640 * 640);  // sigma blurs at L1 res
  float* b150 = alloc((size_t)NPLANE * 640 * 640);
  float* b300 = alloc((size_t)NPLANE * 640 * 640);
  float* tA = alloc((size_t)NPLANE * 320 * 320);    // ping-pong (<=320^2)
  float* tB = alloc((size_t)NPLANE * 320 * 320);

  auto pyr = [&](const float* s, float* d, int W, float sc, float bi) {
    int outW = (W + 1) / 2;
    dim3 grid((outW + 15) / 16, (outW + 15) / 16, NPLANE);
    pyrdown_wmma<<<grid, 32, 0, stream>>>(s, d, W, outW, sc, bi);
  };
  auto gs = [&](const float* s, float* d, int w, int k, float sg) {
    int tot = NPLANE * w * w;
    gauss_small<<<(tot + 255) / 256, 256, 0, stream>>>(s, d, w, k, sg);
  };
  auto up = [&](const float* s, float* d, int w) {
    int tot = NPLANE * 4 * w * w;
    upsample2x<<<(tot + 255) / 256, 256, 0, stream>>>(s, d, w);
  };

  // Shared pyramid (computed once for all three sigmas); fuse img=255x+1 at L0.
  pyr(x,  p1, 1280, 255.0f, 1.0f);
  pyr(p1, p2, 640, 1.0f, 0.0f);
  pyr(p2, p3, 320, 1.0f, 0.0f);
  pyr(p3, p4, 160, 1.0f, 0.0f);
  pyr(p4, p5, 80, 1.0f, 0.0f);
  pyr(p5, p6, 40, 1.0f, 0.0f);
  pyr(p6, p7, 20, 1.0f, 0.0f);
  pyr(p7, p8, 10, 1.0f, 0.0f);

  // sigma=30: blur 7x7 (s=1.4) at L5 (40^2), 4 upsamples -> 640^2
  gs(p5, tA, 40, 7, 1.4f);
  up(tA, tB, 40); up(tB, tA, 80); up(tA, tB, 160); up(tB, b30, 320);

  // sigma=150: blur 9x9 (s=1.7) at L7 (10^2), 6 upsamples -> 640^2
  gs(p7, tA, 10, 9, 1.7f);
  up(tA, tB, 10); up(tB, tA, 20); up(tA, tB, 40);
  up(tB, tA, 80); up(tA, tB, 160); up(tB, b150, 320);

  // sigma=300: blur 9x9 (s=1.7) at L8 (5^2), 7 upsamples -> 640^2
  gs(p8, tA, 5, 9, 1.7f);
  up(tA, tB, 5); up(tB, tA, 10); up(tA, tB, 20); up(tB, tA, 40);
  up(tA, tB, 80); up(tB, tA, 160); up(tA, b300, 320);

  // Fused retinex + color restoration with inline final 2x bilerp.
  int tot = 8 * HFULL * HFULL;
  final_msr<<<(tot + 255) / 256, 256, 0, stream>>>(x, b30, b150, b300, out);
}